// lossFuncTask_84258668413461
// MI455X (gfx1250) — compile-verified
//
#include <hip/hip_runtime.h>
#include <cstdint>

// Problem constants (match reference)
constexpr int Bn  = 256;   // students
constexpr int Tn  = 200;   // steps
constexpr int Qn  = 1000;  // questions
constexpr int TM1 = Tn - 1;

typedef __attribute__((ext_vector_type(4))) unsigned int v4u;
typedef __attribute__((ext_vector_type(8))) int          v8i;
typedef __attribute__((ext_vector_type(4))) int          v4i;

__device__ __forceinline__ float sigmoidf_(float x) {
    return 1.0f / (1.0f + expf(-x));
}

// torch BCELoss with log clamped at -100
__device__ __forceinline__ float bce_elem(float p, float af) {
    float logp   = fmaxf(logf(p),     -100.0f);
    float log1mp = fmaxf(log1pf(-p),  -100.0f);
    return -(af * logp + (1.0f - af) * log1mp);
}

// TDM: DMA one contiguous row of `nelem` 4-byte ints from global to LDS.
// D# packing per CDNA5 ISA §8.3/8.4 (1-D tile, no pad, no iterate, no multicast).
__device__ __forceinline__ void tdm_load_row_to_lds(const int* gptr, void* lds_ptr, int nelem) {
    uint64_t ga = (uint64_t)(uintptr_t)gptr;
    uint32_t la = (uint32_t)(uintptr_t)lds_ptr;   // generic LDS ptr: low 32 bits = LDS byte offset
    v4u g0;
    g0[0] = 1u;                                                  // count=1, is_restore=0, no gather
    g0[1] = la;                                                  // lds_addr
    g0[2] = (uint32_t)ga;                                        // global_addr[31:0]
    g0[3] = (uint32_t)((ga >> 32) & 0x01FFFFFFu) | (2u << 30);   // global_addr[56:32] | type=2
    v8i g1;
    g1[0] = 0x00020000;                                  // wg_mask=0, data_size=2 (4B)
    g1[1] = (int)((uint32_t)nelem << 16);                // tensor_dim0[15:0] in [31:16]
    g1[2] = (int)(1u << 16);                             // tensor_dim0 hi=0, tensor_dim1=1
    g1[3] = (int)((uint32_t)nelem << 16);                // tensor_dim1 hi=0, tile_dim0=nelem
    g1[4] = 1;                                           // tile_dim1=1, tile_dim2=0 (unused)
    g1[5] = nelem;                                       // tensor_dim0_stride[31:0]
    g1[6] = (int)((uint32_t)nelem << 16);                // stride0 hi=0, tensor_dim1_stride lo
    g1[7] = 0;
    v4i g2 = {0, 0, 0, 0};
    v4i g3 = {0, 0, 0, 0};
    v8i g4 = {0, 0, 0, 0, 0, 0, 0, 0};   // 6-arg toolchain form: extra group (unused, 1-D tile)
    __builtin_amdgcn_tensor_load_to_lds(g0, g1, g2, g3, g4, /*cpol=*/0);
}

// One wave32 per student. Stages q_idx/correct rows via TDM, then 7x32
// independent dependent-gather chunks, then shuffle reductions.
__global__ __launch_bounds__(32)
void kt_loss_kernel(const float* __restrict__ logit_c,
                    const float* __restrict__ logit_t,
                    const int*   __restrict__ q_idx,
                    const int*   __restrict__ correct,
                    float* __restrict__ p_mean,            // [B, T-1]
                    float* __restrict__ gt,                // [B, T-1]
                    float* __restrict__ loss_per_student)  // [B]
{
    __shared__ int sm_q[Tn];
    __shared__ int sm_a[Tn];

    const int s    = blockIdx.x;
    const int lane = threadIdx.x;

    // Async-tensor staging of this student's index rows (EXEC-independent, one issue per wave)
    tdm_load_row_to_lds(q_idx   + (size_t)s * Tn, sm_q, Tn);
    tdm_load_row_to_lds(correct + (size_t)s * Tn, sm_a, Tn);
    __builtin_amdgcn_s_wait_tensorcnt(0);
    __syncthreads();   // single-wave WG: compiler fence for the TDM-written LDS

    float ec[7], et[7];
    int last_local = -1;
    #pragma unroll
    for (int k = 0; k < 7; ++k) {
        const int i = k * 32 + lane;
        float e_c = 0.0f, e_t = 0.0f;
        if (i < TM1) {
            const int   q  = sm_q[i + 1];
            const float af = (float)sm_a[i + 1];
            const size_t base = ((size_t)s * Tn + (size_t)i) * Qn + (size_t)q;
            // single-use gathered elements -> non-temporal (don't pollute L2)
            const float lc = __builtin_nontemporal_load(logit_c + base);
            const float lt = __builtin_nontemporal_load(logit_t + base);
            const float pc = sigmoidf_(lc);
            const float pt = sigmoidf_(lt);
            const size_t o = (size_t)s * TM1 + (size_t)i;
            p_mean[o] = 0.5f * (pc + pt);
            gt[o]     = af;
            e_c = bce_elem(pc, af);
            e_t = bce_elem(pt, af);
            if (pc > 0.0f) last_local = i;   // monotone i -> per-lane max valid index
        }
        ec[k] = e_c;
        et[k] = e_t;
    }

    // wave32 max-reduce: last valid index (reference: T-2 - argmax(reversed valid))
    int last = last_local;
    #pragma unroll
    for (int m = 16; m >= 1; m >>= 1)
        last = max(last, __shfl_xor(last, m, 32));
    if (last < 0) last = TM1 - 1;            // no valid -> argmax==0 -> last = T-2
    const float cnt = (float)(last + 1);

    // masked sums
    float sc = 0.0f, st = 0.0f;
    #pragma unroll
    for (int k = 0; k < 7; ++k) {
        const int i = k * 32 + lane;
        if (i < TM1 && i <= last) { sc += ec[k]; st += et[k]; }
    }
    #pragma unroll
    for (int m = 16; m >= 1; m >>= 1) {
        sc += __shfl_xor(sc, m, 32);
        st += __shfl_xor(st, m, 32);
    }

    if (lane == 0)
        loss_per_student[s] = st / cnt + sc / cnt;   // bce_mean(pt) + bce_mean(pc)
}

// Deterministic fixed-order sum of 256 per-student losses (no float atomics).
__global__ __launch_bounds__(32)
void kt_loss_finalize(const float* __restrict__ loss_per_student,
                      float* __restrict__ out_loss)
{
    const int lane = threadIdx.x;
    float acc = 0.0f;
    #pragma unroll
    for (int k = 0; k < Bn / 32; ++k)
        acc += loss_per_student[k * 32 + lane];
    #pragma unroll
    for (int m = 16; m >= 1; m >>= 1)
        acc += __shfl_xor(acc, m, 32);
    if (lane == 0) out_loss[0] = acc;
}

extern "C" void kernel_launch(void* const* d_in, const int* in_sizes, int n_in,
                              void* d_out, int out_size, void* d_ws, size_t ws_size,
                              hipStream_t stream) {
    const float* logit_c = (const float*)d_in[0];
    const float* logit_t = (const float*)d_in[1];
    const int*   q_idx   = (const int*)d_in[2];
    const int*   correct = (const int*)d_in[3];

    float* out    = (float*)d_out;
    float* loss   = out;                     // [1]
    float* p_mean = out + 1;                 // [B*(T-1)]
    float* gt     = out + 1 + Bn * TM1;      // [B*(T-1)]
    float* ws     = (float*)d_ws;            // [B] per-student losses (1 KB)

    kt_loss_kernel<<<Bn, 32, 0, stream>>>(logit_c, logit_t, q_idx, correct, p_mean, gt, ws);
    kt_loss_finalize<<<1, 32, 0, stream>>>(ws, loss);
}